// ContrastiveLoss_66417374265667
// MI455X (gfx1250) — compile-verified
//
#include <hip/hip_runtime.h>
#include <hip/hip_bf16.h>
#include <math.h>

// ---------------- problem constants (from reference) ----------------
#define BT 64        // captions
#define BI 64        // images
#define RR 36        // regions
#define LL 50        // words
#define DD 1024      // feature dim
#define MROWS (BT * LL)   // 3200
#define NCOLS (BI * RR)   // 2304
#define LAMBDA_SM 9.0f
#define MARGIN 0.2f
#define EPSV 1e-8f

#define KCHUNK 64                 // K elements staged per TDM transfer
#define NKCHUNK (DD / KCHUNK)     // 16 chunks

typedef float v2f __attribute__((ext_vector_type(2)));
typedef float v8f __attribute__((ext_vector_type(8)));
typedef unsigned int tdm_v4u __attribute__((ext_vector_type(4)));
typedef int tdm_v8i __attribute__((ext_vector_type(8)));
typedef int tdm_v4i __attribute__((ext_vector_type(4)));

// -------------------------------------------------------------------
// TDM descriptor build + issue: load a [64 rows x KCHUNK floats] tile of
// im (row stride DD floats) into LDS.  D# packing per CDNA5 ISA ch.8.
// Group0: count=1 | lds_addr | global_addr[56:0] | type=2.
// Group1: data_size=4B, tensor_dim0=DD, tensor_dim1=NCOLS,
//         tile_dim0=KCHUNK, tile_dim1=64, tensor_dim0_stride=DD.
// This toolchain exposes the 6-arg builtin:
//   (uint32x4 g0, int32x8 g1, int32x4 g2, int32x4 g3, int32x8 g4, i32 cpol)
// -------------------------------------------------------------------
__device__ __forceinline__ void tdm_load_b_tile(const float* im, int n0, int kb,
                                                void* lds_dst) {
  const unsigned long long ga =
      (unsigned long long)(uintptr_t)(im + (size_t)n0 * DD + kb);
  const unsigned int lds = (unsigned int)(uintptr_t)lds_dst;  // LDS offset
  tdm_v4u g0;
  g0[0] = 1u;                                   // count=1, user desc
  g0[1] = lds;                                  // lds_addr (bits 63:32)
  g0[2] = (unsigned int)ga;                     // global_addr[31:0]
  g0[3] = (unsigned int)(ga >> 32) | (2u << 30);// global_addr[56:32] | type=2
  tdm_v8i g1;
  g1[0] = (int)(2u << 16);                      // data_size=2 (4 bytes)
  g1[1] = (int)((unsigned)DD << 16);            // tensor_dim0[15:0] @ bits 63:48
  g1[2] = (int)((unsigned)NCOLS << 16);         // tensor_dim1[15:0] @ bits 95:80
  g1[3] = (int)((unsigned)KCHUNK << 16);        // tile_dim0 @ bits 127:112
  g1[4] = 64;                                   // tile_dim1 @ bits 143:128
  g1[5] = DD;                                   // tensor_dim0_stride[31:0]
  g1[6] = 0;
  g1[7] = 0;
  tdm_v4i z4 = {0, 0, 0, 0};
  tdm_v8i z8 = {0, 0, 0, 0, 0, 0, 0, 0};
  __builtin_amdgcn_tensor_load_to_lds(g0, g1, z4, z4, z8, 0);
}

// -------------------------------------------------------------------
// Kernel 1: g[m, n] = sum_d s_flat[m, d] * im_flat[n, d]
// M=3200, N=2304, K=1024.  Workgroup = 8 waves = 128x64 output block.
// B panel (64 im rows) staged to LDS by the Tensor Data Mover, double
// buffered; each wave computes a 16x64 strip with V_WMMA_F32_16X16X4_F32
// reading A from global (float2/lane) and B from LDS (ds_load_b64).
// -------------------------------------------------------------------
__global__ __launch_bounds__(256) void cl_gemm_g(const float* __restrict__ s,
                                                 const float* __restrict__ im,
                                                 float* __restrict__ g) {
  __shared__ float bpanel[2][64 * KCHUNK];      // 2 x 16 KB

  const int wv   = threadIdx.x >> 5;            // wave in block: 0..7
  const int lane = threadIdx.x & 31;
  const int mgrp = blockIdx.x / 36;             // 0..24  (8 M-tiles each)
  const int nsup = blockIdx.x % 36;             // 0..35  (64 cols each)
  const int mtile = mgrp * 8 + wv;              // 0..199
  const int m0 = mtile * 16;
  const int n0 = nsup * 64;
  const int ml = lane & 15;
  const int hi = lane >> 4;                     // 0 or 1
  const int nl = lane & 15;

  const float* __restrict__ srow = s + (size_t)(m0 + ml) * DD + 2 * hi;

  v8f acc0 = {}, acc1 = {}, acc2 = {}, acc3 = {};

  // prologue: wave 0 kicks off chunk 0 via the TDM
  if (threadIdx.x < 32) tdm_load_b_tile(im, n0, 0, &bpanel[0][0]);

  for (int c = 0; c < NKCHUNK; ++c) {
    const int kb = c * KCHUNK;
    if (threadIdx.x < 32) {
      if (c + 1 < NKCHUNK) {
        tdm_load_b_tile(im, n0, kb + KCHUNK, &bpanel[(c + 1) & 1][0]);
        __builtin_amdgcn_s_wait_tensorcnt(1);   // chunk c landed
      } else {
        __builtin_amdgcn_s_wait_tensorcnt(0);
      }
    }
    // prefetch next A chunk while we crunch this one
    if (c + 1 < NKCHUNK) __builtin_prefetch(srow + kb + KCHUNK, 0, 1);
    __syncthreads();                            // publish buffer to all waves

    const float* bp = &bpanel[c & 1][0];
#pragma unroll 8
    for (int kk = 0; kk < KCHUNK; kk += 4) {
      v2f a  = *(const v2f*)(srow + kb + kk);
      v2f b0 = *(const v2f*)(bp + (nl     ) * KCHUNK + kk + 2 * hi);
      v2f b1 = *(const v2f*)(bp + (nl + 16) * KCHUNK + kk + 2 * hi);
      v2f b2 = *(const v2f*)(bp + (nl + 32) * KCHUNK + kk + 2 * hi);
      v2f b3 = *(const v2f*)(bp + (nl + 48) * KCHUNK + kk + 2 * hi);
      // 8 args: (neg_a, A, neg_b, B, c_mod, C, reuse_a, reuse_b)
      acc0 = __builtin_amdgcn_wmma_f32_16x16x4_f32(false, a, false, b0, (short)0, acc0, false, false);
      acc1 = __builtin_amdgcn_wmma_f32_16x16x4_f32(false, a, false, b1, (short)0, acc1, false, false);
      acc2 = __builtin_amdgcn_wmma_f32_16x16x4_f32(false, a, false, b2, (short)0, acc2, false, false);
      acc3 = __builtin_amdgcn_wmma_f32_16x16x4_f32(false, a, false, b3, (short)0, acc3, false, false);
    }
    __syncthreads();                            // buffer reuse fence
  }

  // D layout (16x16 f32, 8 vgprs): vgpr r -> row (r + 8*hi), col = lane&15
#pragma unroll
  for (int r = 0; r < 8; ++r) {
    const int row = m0 + r + 8 * hi;
    float* grow = g + (size_t)row * NCOLS + n0 + nl;
    grow[0]  = acc0[r];
    grow[16] = acc1[r];
    grow[32] = acc2[r];
    grow[48] = acc3[r];
  }
}

// -------------------------------------------------------------------
// Kernel 2: gram[j, r, q] = dot(im[j,r,:], im[j,q,:])   (64 x 36 x 36)
// -------------------------------------------------------------------
__global__ __launch_bounds__(256) void cl_gram(const float* __restrict__ im,
                                               float* __restrict__ gram) {
  const int idx = blockIdx.x * 256 + threadIdx.x;
  if (idx >= BI * RR * RR) return;
  const int q = idx % RR;
  const int r = (idx / RR) % RR;
  const int j = idx / (RR * RR);
  const float4* pr = (const float4*)(im + ((size_t)j * RR + r) * DD);
  const float4* pq = (const float4*)(im + ((size_t)j * RR + q) * DD);
  float acc = 0.f;
#pragma unroll 4
  for (int d = 0; d < DD / 4; ++d) {
    float4 x = pr[d], y = pq[d];
    acc += x.x * y.x + x.y * y.y + x.z * y.z + x.w * y.w;
  }
  gram[idx] = acc;
}

// -------------------------------------------------------------------
// Kernel 3: row L2 norms of s (3200 rows) and im (2304 rows)
// -------------------------------------------------------------------
__global__ __launch_bounds__(256) void cl_norms(const float* __restrict__ s,
                                                const float* __restrict__ im,
                                                float* __restrict__ ns,
                                                float* __restrict__ nim) {
  const int idx = blockIdx.x * 256 + threadIdx.x;
  if (idx >= MROWS + NCOLS) return;
  const float4* row;
  float* out;
  if (idx < MROWS) { row = (const float4*)(s + (size_t)idx * DD);            out = ns  + idx; }
  else             { row = (const float4*)(im + (size_t)(idx - MROWS) * DD); out = nim + idx - MROWS; }
  float acc = 0.f;
#pragma unroll 4
  for (int d = 0; d < DD / 4; ++d) {
    float4 x = row[d];
    acc += x.x * x.x + x.y * x.y + x.z * x.z + x.w * x.w;
  }
  *out = sqrtf(acc);
}

// -------------------------------------------------------------------
// Kernel 4: per-(caption i, image j) attention pipeline -> scores[i,j]
// -------------------------------------------------------------------
__global__ __launch_bounds__(256) void cl_pair(const float* __restrict__ g,
                                               const float* __restrict__ gram,
                                               const float* __restrict__ ns,
                                               const float* __restrict__ nim,
                                               const int* __restrict__ img_ent,
                                               const int* __restrict__ cap_ent,
                                               const int* __restrict__ s_l,
                                               float* __restrict__ scores) {
  const int i = blockIdx.x >> 6;   // caption
  const int j = blockIdx.x & 63;   // image
  const int tid = threadIdx.x;

  __shared__ float gt[LL * RR];      // raw dot tile  [50,36]
  __shared__ float at[LL * RR];      // attn / a      [50,36]
  __shared__ float gj[RR * RR];      // gram[j]       [36,36]
  __shared__ float norm_r[RR];
  __shared__ float cosv[LL];
  __shared__ float red[256];
  __shared__ float latent_s;

  const int len  = s_l[i];
  const int nmin = len < LL ? len : LL;

  for (int idx = tid; idx < LL * RR; idx += 256) {
    const int w = idx / RR, r = idx % RR;
    gt[idx] = g[(size_t)(i * LL + w) * NCOLS + j * RR + r];
  }
  for (int idx = tid; idx < RR * RR; idx += 256)
    gj[idx] = gram[(size_t)j * RR * RR + idx];
  __syncthreads();

  // attn = leaky_relu(g, 0.1), zero padded words
  for (int idx = tid; idx < LL * RR; idx += 256) {
    const int w = idx / RR;
    float v = gt[idx];
    v = v > 0.f ? v : 0.1f * v;
    if (w >= len) v = 0.f;
    at[idx] = v;
  }
  __syncthreads();

  // l2 norm over word axis per region r
  if (tid < RR) {
    float acc = 0.f;
    for (int w = 0; w < LL; ++w) { float v = at[w * RR + tid]; acc += v * v; }
    norm_r[tid] = sqrtf(acc) + EPSV;
  }
  __syncthreads();

  // per word: softmax over regions, filter, dot, |wC|, cosine
  if (tid < LL) {
    const int w = tid;
    float vals[RR];
    float mx = -1e30f;
#pragma unroll
    for (int r = 0; r < RR; ++r) {
      float v = (at[w * RR + r] / norm_r[r]) * LAMBDA_SM;
      vals[r] = v;
      mx = fmaxf(mx, v);
    }
    float sum = 0.f;
#pragma unroll
    for (int r = 0; r < RR; ++r) { float e = __expf(vals[r] - mx); vals[r] = e; sum += e; }
    const float inv = 1.f / sum;
    float dot_swc = 0.f;
#pragma unroll
    for (int r = 0; r < RR; ++r) {
      float aa = vals[r] * inv;
      aa = (aa > (1.0f / (float)RR)) ? aa : 0.f;  // attn_filter
      at[w * RR + r] = aa;
      dot_swc += aa * gt[w * RR + r];
    }
    float wcsq = 0.f;
    for (int q = 0; q < RR; ++q) {
      float tq = 0.f;
#pragma unroll 4
      for (int r = 0; r < RR; ++r) tq += at[w * RR + r] * gj[r * RR + q];
      wcsq += tq * at[w * RR + q];
    }
    const float wc_norm = sqrtf(fmaxf(wcsq, 1e-24f));
    const float denom = fmaxf(ns[i * LL + w] * wc_norm, EPSV);
    float c = dot_swc / denom;
    if (w >= len) c = -1e30f;
    cosv[w] = c;
  }
  __syncthreads();

  // latent: mean of top-k cosines
  if (tid == 0) {
    const int k = len - len / 3;
    float loc[LL];
#pragma unroll 2
    for (int w = 0; w < LL; ++w) loc[w] = cosv[w];
    float acc = 0.f;
    for (int t = 0; t < k; ++t) {
      float mx = -1e30f; int mi = 0;
      for (int w = 0; w < LL; ++w)
        if (loc[w] > mx) { mx = loc[w]; mi = w; }
      acc += mx;
      loc[mi] = -1e30f;
    }
    latent_s = acc / (float)k;
  }

  // direct entity-matched score (block reduce)
  float part = 0.f;
  for (int idx = tid; idx < LL * RR; idx += 256) {
    const int w = idx / RR, r = idx % RR;
    const int ce = cap_ent[i * LL + w];
    if (ce != 0 && w < nmin && ce == img_ent[j * RR + r]) {
      const float denom = fmaxf(ns[i * LL + w] * nim[j * RR + r], EPSV);
      part += gt[idx] / denom;
    }
  }
  red[tid] = part;
  __syncthreads();
#pragma unroll
  for (int sft = 128; sft > 0; sft >>= 1) {
    if (tid < sft) red[tid] += red[tid + sft];
    __syncthreads();
  }
  if (tid == 0)
    scores[i * BI + j] = latent_s + red[0] / (float)nmin;
}

// -------------------------------------------------------------------
// Kernel 5: max-violation hinge loss over 64x64 scores
// -------------------------------------------------------------------
__global__ __launch_bounds__(64) void cl_loss(const float* __restrict__ scores,
                                              float* __restrict__ out) {
  __shared__ float diag[BT];
  __shared__ float red[BT];
  const int t = threadIdx.x;
  diag[t] = scores[t * BI + t];
  __syncthreads();
  const float dt = diag[t];
  float cs = 0.f, ci = 0.f;
  for (int j = 0; j < BI; ++j) {
    if (j == t) continue;
    cs = fmaxf(cs, fmaxf(MARGIN + scores[t * BI + j] - dt, 0.f)); // cost_s row t
    ci = fmaxf(ci, fmaxf(MARGIN + scores[j * BI + t] - dt, 0.f)); // cost_im col t
  }
  red[t] = cs + ci;
  __syncthreads();
#pragma unroll
  for (int sft = 32; sft > 0; sft >>= 1) {
    if (t < sft) red[t] += red[t + sft];
    __syncthreads();
  }
  if (t == 0) out[0] = red[0];
}

// -------------------------------------------------------------------
extern "C" void kernel_launch(void* const* d_in, const int* in_sizes, int n_in,
                              void* d_out, int out_size, void* d_ws, size_t ws_size,
                              hipStream_t stream) {
  (void)in_sizes; (void)n_in; (void)out_size; (void)ws_size;
  const float* im      = (const float*)d_in[0];   // [64,36,1024]
  const float* s       = (const float*)d_in[1];   // [64,50,1024]
  const int*   img_ent = (const int*)  d_in[2];   // [64,36]
  const int*   cap_ent = (const int*)  d_in[3];   // [64,50]
  const int*   s_l     = (const int*)  d_in[4];   // [64]
  float* out = (float*)d_out;

  float* ws     = (float*)d_ws;
  float* g      = ws;                                // 3200*2304
  float* gram   = g    + (size_t)MROWS * NCOLS;      // 64*36*36
  float* ns     = gram + (size_t)BI * RR * RR;       // 3200
  float* nim    = ns   + MROWS;                      // 2304
  float* scores = nim  + NCOLS;                      // 64*64

  // 25 M-groups (128 rows) * 36 N-supertiles = 900 blocks of 8 waves
  cl_gemm_g<<<900, 256, 0, stream>>>(s, im, g);
  cl_gram<<<(BI * RR * RR + 255) / 256, 256, 0, stream>>>(im, gram);
  cl_norms<<<(MROWS + NCOLS + 255) / 256, 256, 0, stream>>>(s, im, ns, nim);
  cl_pair<<<BT * BI, 256, 0, stream>>>(g, gram, ns, nim, img_ent, cap_ent, s_l, scores);
  cl_loss<<<1, 64, 0, stream>>>(scores, out);
}